// Prompt_31679678775553
// MI455X (gfx1250) — compile-verified
//
#include <hip/hip_runtime.h>
#include <hip/hip_bf16.h>

// ---------------- problem constants ----------------
#define BATCH      1024
#define N_NODES    100
#define KEY_DIM    640          // 5*128
#define POOL       128
#define TOPK       4
#define PROMPT_DIM 2304         // 3*6*128

// d_out layout (flat, return order)
#define OUT_PROMPT_OFF 0
#define OUT_PROMPT_N   (BATCH * TOPK * PROMPT_DIM)      // 9,437,184
#define OUT_RSIM_OFF   (OUT_PROMPT_OFF + OUT_PROMPT_N)  // 9,437,184
#define OUT_SIM_OFF    (OUT_RSIM_OFF + 1)               // 9,437,185
#define OUT_SIM_N      (BATCH * POOL)                   // 131,072
#define OUT_USAGE_OFF  (OUT_SIM_OFF + OUT_SIM_N)        // 9,568,257

// workspace layout (floats / ints)
#define WS_XNORM_OFF   0                                // 1024*640 f32
#define WS_KNORM_OFF   (WS_XNORM_OFF + BATCH * KEY_DIM) // 128*640 f32
#define WS_PARTIAL_OFF (WS_KNORM_OFF + POOL * KEY_DIM)  // 1024 f32
#define WS_USAGE_OFF   (WS_PARTIAL_OFF + BATCH)         // 128 i32 (reinterpret)

typedef __attribute__((ext_vector_type(2))) float v2f;
typedef __attribute__((ext_vector_type(8))) float v8f;

// ---------------- k0: init usage scratch ----------------
__global__ void k0_init(int* __restrict__ usage) {
    if (threadIdx.x < POOL) usage[threadIdx.x] = 0;
}

// ---------------- k1: mean over nodes + L2 normalize ----------------
// one block per batch row; threads 0..159 each own a float4 column slice
__global__ __launch_bounds__(256)
void k1_mean_norm(const float* __restrict__ x_embed, float* __restrict__ x_norm) {
    __shared__ float sdata[256];
    const int b = blockIdx.x;
    const int t = threadIdx.x;

    float4 acc = {0.f, 0.f, 0.f, 0.f};
    if (t < KEY_DIM / 4) {
        const float4* row = (const float4*)(x_embed + (size_t)b * N_NODES * KEY_DIM);
        const int nf4 = KEY_DIM / 4; // 160
        #pragma unroll 4
        for (int n = 0; n < N_NODES; ++n) {
            float4 v = row[(size_t)n * nf4 + t];
            acc.x += v.x; acc.y += v.y; acc.z += v.z; acc.w += v.w;
        }
        const float inv_n = 1.0f / (float)N_NODES;
        acc.x *= inv_n; acc.y *= inv_n; acc.z *= inv_n; acc.w *= inv_n;
    }
    float ss = acc.x * acc.x + acc.y * acc.y + acc.z * acc.z + acc.w * acc.w;
    sdata[t] = (t < KEY_DIM / 4) ? ss : 0.f;
    __syncthreads();
    for (int s = 128; s > 0; s >>= 1) {
        if (t < s) sdata[t] += sdata[t + s];
        __syncthreads();
    }
    const float rs = rsqrtf(fmaxf(sdata[0], 1e-12f));
    if (t < KEY_DIM / 4) {
        float4 o = {acc.x * rs, acc.y * rs, acc.z * rs, acc.w * rs};
        ((float4*)(x_norm + (size_t)b * KEY_DIM))[t] = o;
    }
}

// ---------------- k2: L2 normalize prompt keys ----------------
__global__ __launch_bounds__(256)
void k2_key_norm(const float* __restrict__ prompt_key, float* __restrict__ key_norm) {
    __shared__ float sdata[256];
    const int p = blockIdx.x;
    const int t = threadIdx.x;
    float4 v = {0.f, 0.f, 0.f, 0.f};
    if (t < KEY_DIM / 4) v = ((const float4*)(prompt_key + (size_t)p * KEY_DIM))[t];
    float ss = v.x * v.x + v.y * v.y + v.z * v.z + v.w * v.w;
    sdata[t] = (t < KEY_DIM / 4) ? ss : 0.f;
    __syncthreads();
    for (int s = 128; s > 0; s >>= 1) {
        if (t < s) sdata[t] += sdata[t + s];
        __syncthreads();
    }
    const float rs = rsqrtf(fmaxf(sdata[0], 1e-12f));
    if (t < KEY_DIM / 4) {
        float4 o = {v.x * rs, v.y * rs, v.z * rs, v.w * rs};
        ((float4*)(key_norm + (size_t)p * KEY_DIM))[t] = o;
    }
}

// ---------------- k3: similarity GEMM via V_WMMA_F32_16X16X4_F32 ----------------
// 512 tiles of 16x16 (64 batch-tiles x 8 pool-tiles); one wave32 per tile,
// 4 waves per block -> 128 blocks. K = 640 in steps of 4.
__global__ __launch_bounds__(128)
void k3_sim_wmma(const float* __restrict__ xn, const float* __restrict__ kn,
                 float* __restrict__ sim) {
    const int lane = threadIdx.x & 31;
    const int wave = threadIdx.x >> 5;
    const int tile = blockIdx.x * 4 + wave;      // 0..511
    const int tm = tile >> 3;                    // batch tile 0..63
    const int tn = tile & 7;                     // pool  tile 0..7

    const int l16  = lane & 15;
    const int khi  = (lane >> 4) << 1;           // lanes 16..31 take K+2 pair (ISA A/B layout)
    const int m    = tm * 16 + l16;              // A row (batch)
    const int n    = tn * 16 + l16;              // B col (pool)

    const float* arow = xn + (size_t)m * KEY_DIM + khi;
    const float* brow = kn + (size_t)n * KEY_DIM + khi;

    v8f c = {};
    #pragma unroll 8
    for (int k = 0; k < KEY_DIM; k += 4) {
        v2f a, b;
        a.x = arow[k];     a.y = arow[k + 1];
        b.x = brow[k];     b.y = brow[k + 1];
        // D = A(16x4) * B(4x16) + C   (f32 everywhere)
        c = __builtin_amdgcn_wmma_f32_16x16x4_f32(
                /*neg_a=*/false, a, /*neg_b=*/false, b,
                /*c_mod=*/(short)0, c, /*reuse_a=*/false, /*reuse_b=*/false);
    }

    // C/D layout: VGPR j -> M = j + (lane>=16 ? 8 : 0), N = lane&15
    const int rowBase = tm * 16 + ((lane >> 4) << 3);
    const int col     = tn * 16 + l16;
    #pragma unroll
    for (int j = 0; j < 8; ++j)
        sim[(size_t)(rowBase + j) * POOL + col] = c[j];
}

// ---------------- k4: top-4, prompt gather, usage, partial reduce_sim ----------------
// one block (256 threads) per batch row
__global__ __launch_bounds__(256)
void k4_topk_gather(const float* __restrict__ sim, const float* __restrict__ prompt,
                    float* __restrict__ out_prompt, float* __restrict__ partial,
                    int* __restrict__ usage) {
    __shared__ int   sIdx[TOPK];
    __shared__ float sVal[TOPK];

    const int b   = blockIdx.x;
    const int tid = threadIdx.x;

    if (tid < 32) {
        const int lane = tid;
        // each lane owns cols lane, lane+32, lane+64, lane+96
        float v0 = sim[(size_t)b * POOL + lane];
        float v1 = sim[(size_t)b * POOL + lane + 32];
        float v2 = sim[(size_t)b * POOL + lane + 64];
        float v3 = sim[(size_t)b * POOL + lane + 96];
        float vals[4] = {v0, v1, v2, v3};
        unsigned alive = 0xF;

        for (int kk = 0; kk < TOPK; ++kk) {
            float lv = -3.4e38f;
            int   li = 0x7FFFFFFF;
            #pragma unroll
            for (int j = 0; j < 4; ++j) {
                if ((alive >> j) & 1) {
                    float v = vals[j];
                    int   c = lane + 32 * j;
                    if (v > lv || (v == lv && c < li)) { lv = v; li = c; }
                }
            }
            // wave32 argmax, tie -> smaller index (matches top_k stability)
            #pragma unroll
            for (int off = 16; off > 0; off >>= 1) {
                float ov = __shfl_xor(lv, off, 32);
                int   oi = __shfl_xor(li, off, 32);
                if (ov > lv || (ov == lv && oi < li)) { lv = ov; li = oi; }
            }
            if ((li & 31) == lane) alive &= ~(1u << (li >> 5)); // owner retires column
            if (lane == 0) { sIdx[kk] = li; sVal[kk] = lv; }
        }
    }
    __syncthreads();

    if (tid == 0)
        partial[b] = sVal[0] + sVal[1] + sVal[2] + sVal[3];
    if (tid < TOPK)
        atomicAdd(&usage[sIdx[tid]], 1);

    // gather: out[b, kk, :] = prompt[sIdx[kk], :]   (float4 copies)
    const int nf4 = PROMPT_DIM / 4; // 576
    float4* dst = (float4*)(out_prompt + (size_t)b * TOPK * PROMPT_DIM);
    for (int i = tid; i < TOPK * nf4; i += 256) {
        const int kk = i / nf4;
        const int c4 = i - kk * nf4;
        const float4* src = (const float4*)(prompt + (size_t)sIdx[kk] * PROMPT_DIM);
        dst[(size_t)kk * nf4 + c4] = src[c4];
    }
}

// ---------------- k5: deterministic reduce_sim + usage -> float ----------------
__global__ __launch_bounds__(256)
void k5_finalize(const float* __restrict__ partial, const int* __restrict__ usage,
                 float* __restrict__ d_out) {
    __shared__ float sdata[256];
    const int t = threadIdx.x;
    float s = 0.f;
    #pragma unroll
    for (int i = 0; i < BATCH / 256; ++i) s += partial[t + i * 256];
    sdata[t] = s;
    __syncthreads();
    for (int st = 128; st > 0; st >>= 1) {
        if (t < st) sdata[t] += sdata[t + st];
        __syncthreads();
    }
    if (t == 0) d_out[OUT_RSIM_OFF] = sdata[0] / (float)BATCH;
    if (t < POOL) d_out[OUT_USAGE_OFF + t] = (float)usage[t];
}

// ---------------- launcher ----------------
extern "C" void kernel_launch(void* const* d_in, const int* in_sizes, int n_in,
                              void* d_out, int out_size, void* d_ws, size_t ws_size,
                              hipStream_t stream) {
    const float* x_embed    = (const float*)d_in[0]; // [1024,100,640]
    const float* prompt     = (const float*)d_in[1]; // [128,1,2304]
    const float* prompt_key = (const float*)d_in[2]; // [128,640]
    float* out = (float*)d_out;

    float* ws_f   = (float*)d_ws;
    float* xnorm  = ws_f + WS_XNORM_OFF;
    float* knorm  = ws_f + WS_KNORM_OFF;
    float* part   = ws_f + WS_PARTIAL_OFF;
    int*   usage  = (int*)(ws_f + WS_USAGE_OFF);

    k0_init<<<1, 128, 0, stream>>>(usage);
    k1_mean_norm<<<BATCH, 256, 0, stream>>>(x_embed, xnorm);
    k2_key_norm<<<POOL, 256, 0, stream>>>(prompt_key, knorm);
    k3_sim_wmma<<<128, 128, 0, stream>>>(xnorm, knorm, out + OUT_SIM_OFF);
    k4_topk_gather<<<BATCH, 256, 0, stream>>>(out + OUT_SIM_OFF, prompt,
                                              out + OUT_PROMPT_OFF, part, usage);
    k5_finalize<<<1, 256, 0, stream>>>(part, usage, out);
}